// GCN_50680614093670
// MI455X (gfx1250) — compile-verified
//
#include <hip/hip_runtime.h>

typedef __attribute__((ext_vector_type(2))) float v2f;
typedef __attribute__((ext_vector_type(8))) float v8f;

#define NNODES 50000
#define HDIM   96
#define LDB    104   // padded LDS row stride (floats): 2*104 % 64 == 16 -> conflict-free halves

// ---------------------------------------------------------------------------
// Degree / norm kernels
// ---------------------------------------------------------------------------
__global__ void deg_init_kernel(float* __restrict__ deg, int n) {
    int i = blockIdx.x * blockDim.x + threadIdx.x;
    if (i < n) deg[i] = 1.0f;  // self-loop contributes 1
}

__global__ void deg_count_kernel(const int* __restrict__ dst, float* __restrict__ deg, int e) {
    int i = blockIdx.x * blockDim.x + threadIdx.x;
    if (i < e) atomicAdd(&deg[dst[i]], 1.0f);
}

__global__ void dinv_kernel(float* __restrict__ deg, int n) {
    int i = blockIdx.x * blockDim.x + threadIdx.x;
    if (i < n) deg[i] = rsqrtf(deg[i]);   // deg >= 1 always; in place -> dinv
}

// ---------------------------------------------------------------------------
// WMMA f32 GEMM: C[M x 96] = op(A[M x K]) @ B[K x 96] (+ bias), op = optional ReLU.
// - B staged once per block into padded LDS (b128 loads -> ds_store_b128).
// - One wave per 16-row panel, 6 accumulators across the full 96-col width:
//   A fragment loaded once per K-step and shared by 6 WMMAs -> A read exactly once.
// Requires M%16==0, K%4==0.
//
// f32 fragment layouts (ISA 7.12.2):
//   A 16x4 : lanes 0-15 -> M=lane; VGPR0 = K=0 / K=2 (half-wave), VGPR1 = K=1 / K=3
//   B 4x16 : lanes 0-15 -> N=lane; same K split
//   C 16x16: lane -> N=(lane&15); VGPR j -> M = j + 8*(lane>>4)
// ---------------------------------------------------------------------------
template <bool RELU_A, bool HAS_BIAS>
__global__ __launch_bounds__(256)
void wmma_gemm_f32(const float* __restrict__ A, const float* __restrict__ B,
                   const float* __restrict__ bias, float* __restrict__ C,
                   int M, int K) {
    __shared__ float Bs[HDIM * LDB];     // max K=96 rows x 104 floats = 39.9 KB

    const int tid = threadIdx.x;

    // Cooperative B staging: K*24 float4s, row-major global -> padded LDS rows
    const int nf4 = K * 24;
    for (int g = tid; g < nf4; g += 256) {
        int row = g / 24;
        int c4  = g - row * 24;
        float4 v = ((const float4*)B)[g];
        *(float4*)&Bs[row * LDB + c4 * 4] = v;
    }
    __syncthreads();

    const int lane = tid & 31;
    const int wave = tid >> 5;
    const int panel = blockIdx.x * 8 + wave;   // uniform per wave
    const int tilesM = M >> 4;
    if (panel >= tilesM) return;               // EXEC stays all-1s per wave

    const int m0 = panel << 4;
    const int r  = lane & 15;                  // row (A) / col (B) within tile
    const int kk = (lane >> 4) << 1;           // 0 or 2: K-pair for this half-wave

    v8f acc[6] = {};
    for (int k0 = 0; k0 < K; k0 += 4) {
        const float* ap = A + (size_t)(m0 + r) * K + (k0 + kk);
        float a0 = ap[0];
        float a1 = ap[1];
        if (RELU_A) { a0 = fmaxf(a0, 0.0f); a1 = fmaxf(a1, 0.0f); }
        v2f af; af.x = a0; af.y = a1;

        const float* bp = &Bs[(k0 + kk) * LDB + r];
#pragma unroll
        for (int t = 0; t < 6; ++t) {
            v2f bf;
            bf.x = bp[t * 16];
            bf.y = bp[t * 16 + LDB];
            acc[t] = __builtin_amdgcn_wmma_f32_16x16x4_f32(
                /*neg_a=*/false, af, /*neg_b=*/false, bf,
                /*c_mod=*/(short)0, acc[t], /*reuse_a=*/false, /*reuse_b=*/false);
        }
    }

    const int rbase = m0 + ((lane >> 4) << 3);
#pragma unroll
    for (int t = 0; t < 6; ++t) {
        const int col = t * 16 + r;
        const float bv = HAS_BIAS ? bias[col] : 0.0f;
#pragma unroll
        for (int j = 0; j < 8; ++j) {
            C[(size_t)(rbase + j) * HDIM + col] = acc[t][j] + bv;
        }
    }
}

// ---------------------------------------------------------------------------
// agg[n][j] = b[j] + hW[n][j] * dinv[n]^2    (bias + self-loop fused), float4
// ---------------------------------------------------------------------------
__global__ void init_agg_kernel(const float4* __restrict__ hW4, const float* __restrict__ dinv,
                                const float4* __restrict__ bias4, float4* __restrict__ agg4,
                                int total4) {
    int idx = blockIdx.x * blockDim.x + threadIdx.x;
    if (idx >= total4) return;
    int n  = idx / 24;
    int jj = idx - n * 24;
    float di = dinv[n];
    float s = di * di;
    float4 h = hW4[idx];
    float4 b = bias4[jj];
    float4 o;
    o.x = b.x + h.x * s;
    o.y = b.y + h.y * s;
    o.z = b.z + h.z * s;
    o.w = b.w + h.w * s;
    agg4[idx] = o;
}

// ---------------------------------------------------------------------------
// Edge scatter: agg[dst] += hW[src] * dinv[src]*dinv[dst]
// One thread per (edge, 4 features): b128 gather + 4 f32 atomics.
// ---------------------------------------------------------------------------
__global__ void scatter_kernel(const float* __restrict__ hW, const int* __restrict__ src,
                               const int* __restrict__ dst, const float* __restrict__ dinv,
                               float* __restrict__ agg, int total4) {
    int idx = blockIdx.x * blockDim.x + threadIdx.x;
    if (idx >= total4) return;
    int e  = idx / 24;
    int jj = idx - e * 24;
    int s = src[e];
    int d = dst[e];
    float norm = dinv[s] * dinv[d];
    float4 h = *(const float4*)(hW + (size_t)s * HDIM + jj * 4);
    float* ap = agg + (size_t)d * HDIM + jj * 4;
    atomicAdd(ap + 0, h.x * norm);
    atomicAdd(ap + 1, h.y * norm);
    atomicAdd(ap + 2, h.z * norm);
    atomicAdd(ap + 3, h.w * norm);
}

// ---------------------------------------------------------------------------
// out[n][o] = b_out[o] + sum_k relu(h[n][k]) * W_out[k][o],   OUT = 4
// ---------------------------------------------------------------------------
__global__ void out_proj_kernel(const float* __restrict__ h, const float* __restrict__ Wout,
                                const float* __restrict__ bout, float* __restrict__ out,
                                int n) {
    int i = blockIdx.x * blockDim.x + threadIdx.x;
    if (i >= n) return;
    float a0 = bout[0], a1 = bout[1], a2 = bout[2], a3 = bout[3];
    const float* hp = h + (size_t)i * HDIM;
#pragma unroll 4
    for (int k = 0; k < HDIM; ++k) {
        float v = fmaxf(hp[k], 0.0f);
        a0 += v * Wout[k * 4 + 0];
        a1 += v * Wout[k * 4 + 1];
        a2 += v * Wout[k * 4 + 2];
        a3 += v * Wout[k * 4 + 3];
    }
    out[(size_t)i * 4 + 0] = a0;
    out[(size_t)i * 4 + 1] = a1;
    out[(size_t)i * 4 + 2] = a2;
    out[(size_t)i * 4 + 3] = a3;
}

// ---------------------------------------------------------------------------
// Launch
// ---------------------------------------------------------------------------
extern "C" void kernel_launch(void* const* d_in, const int* in_sizes, int n_in,
                              void* d_out, int out_size, void* d_ws, size_t ws_size,
                              hipStream_t stream) {
    const float* x      = (const float*)d_in[0];
    const int*   edge   = (const int*)d_in[1];
    const float* W_in   = (const float*)d_in[2];
    const float* b_in   = (const float*)d_in[3];
    const float* conv_W = (const float*)d_in[4];
    const float* conv_b = (const float*)d_in[5];
    const float* W_out  = (const float*)d_in[6];
    const float* b_out  = (const float*)d_in[7];
    float*       out    = (float*)d_out;

    const int N  = NNODES;
    const int E  = in_sizes[1] / 2;
    const int IN = 16;
    const int H  = HDIM;

    const int* src = edge;       // edge_index[0]
    const int* dst = edge + E;   // edge_index[1]

    // workspace layout: dinv[N], then three N*H ping/pong buffers (~58 MB)
    float* ws   = (float*)d_ws;
    float* dinv = ws;
    float* bufA = ws + N;
    float* bufB = bufA + (size_t)N * H;
    float* bufC = bufB + (size_t)N * H;
    (void)ws_size;

    const int B256 = 256;

    // 1) degrees -> dinv (in place)
    deg_init_kernel<<<(N + B256 - 1) / B256, B256, 0, stream>>>(dinv, N);
    deg_count_kernel<<<(E + B256 - 1) / B256, B256, 0, stream>>>(dst, dinv, E);
    dinv_kernel<<<(N + B256 - 1) / B256, B256, 0, stream>>>(dinv, N);

    const int tilesM   = N / 16;                  // 3125 row panels
    const int gemmGrid = (tilesM + 7) / 8;        // 8 waves per block

    // 2) input projection: h0 = x @ W_in + b_in   (K=16)
    wmma_gemm_f32<false, true><<<gemmGrid, B256, 0, stream>>>(x, W_in, b_in, bufA, N, IN);

    // 3) three GCN layers
    float* h   = bufA;
    float* hW  = bufB;
    float* agg = bufC;
    const int nh4 = N * (H / 4);
    const int eh4 = E * (H / 4);

    for (int l = 0; l < 3; ++l) {
        const float* Wl = conv_W + (size_t)l * H * H;
        const float* bl = conv_b + (size_t)l * H;

        // hW = op(h) @ Wl ; ReLU fused for layers > 0 (h0 enters conv0 un-ReLU'd)
        if (l == 0)
            wmma_gemm_f32<false, false><<<gemmGrid, B256, 0, stream>>>(h, Wl, nullptr, hW, N, H);
        else
            wmma_gemm_f32<true, false><<<gemmGrid, B256, 0, stream>>>(h, Wl, nullptr, hW, N, H);

        // agg = bias + self-loop, then scatter edges
        init_agg_kernel<<<(nh4 + B256 - 1) / B256, B256, 0, stream>>>(
            (const float4*)hW, dinv, (const float4*)bl, (float4*)agg, nh4);
        scatter_kernel<<<(eh4 + B256 - 1) / B256, B256, 0, stream>>>(
            hW, src, dst, dinv, agg, eh4);

        // rotate: agg (pre-ReLU h_{l+1}) becomes next h
        float* tmp = h; h = agg; agg = tmp;
    }

    // 4) output projection with fused ReLU
    out_proj_kernel<<<(N + B256 - 1) / B256, B256, 0, stream>>>(h, W_out, b_out, out, N);
}